// GraphAttention_3221225472506
// MI455X (gfx1250) — compile-verified
//
#include <hip/hip_runtime.h>
#include <hip/hip_bf16.h>

typedef __attribute__((ext_vector_type(2))) float v2f;
typedef __attribute__((ext_vector_type(8))) float v8f;

#define N_NODES   50000
#define N_EDGES   800000
#define IN_FEAT   256
#define UNITS     128
#define NEG_SLOPE 0.2f

// ---------------------------------------------------------------------------
// Kernel 1: h = X @ W via V_WMMA_F32_16X16X4_F32 (fp32 matrix pipe, wave32).
// grid.x = N_NODES/16 = 3125 (exact), 256 threads = 8 waves.
// Each wave computes one 16(M) x 16(N) tile; wave w owns columns [16w,16w+16).
// A-tile staged in LDS (padded stride 258 -> conflict-free strided reads).
// ---------------------------------------------------------------------------
__global__ __launch_bounds__(256) void gat_gemm_h(const float* __restrict__ X,
                                                  const float* __restrict__ W,
                                                  float* __restrict__ H)
{
    __shared__ float Atile[16][IN_FEAT + 2];   // stride 258 floats (even -> b64 ok)

    const int n0 = blockIdx.x * 16;

    // Cooperative load: 16 rows x 256 cols, fully coalesced.
    for (int i = threadIdx.x; i < 16 * IN_FEAT; i += 256) {
        const int r = i >> 8;          // / 256
        const int c = i & (IN_FEAT - 1);
        Atile[r][c] = X[(size_t)(n0 + r) * IN_FEAT + c];
    }
    __syncthreads();

    const int wave  = threadIdx.x >> 5;
    const int lane  = threadIdx.x & 31;
    const int m_l   = lane & 15;       // A row / B,C column within tile
    const int khalf = lane >> 4;       // 0: K pair {0,1}; 1: K pair {2,3}
    const int ncol  = wave * 16 + m_l;

    v8f acc = {};
    #pragma unroll 4
    for (int ks = 0; ks < IN_FEAT; ks += 4) {
        const int k0 = ks + 2 * khalf;
        // A fragment: A[m_l][k0], A[m_l][k0+1] (contiguous -> ds_load_b64)
        v2f a = *(const v2f*)&Atile[m_l][k0];
        // B fragment: B[k0][ncol], B[k0+1][ncol]
        v2f b;
        b.x = W[(size_t)k0       * UNITS + ncol];
        b.y = W[(size_t)(k0 + 1) * UNITS + ncol];
        acc = __builtin_amdgcn_wmma_f32_16x16x4_f32(
                  /*neg_a=*/false, a, /*neg_b=*/false, b,
                  /*c_mod=*/(short)0, acc, /*reuse_a=*/false, /*reuse_b=*/false);
    }

    // C/D layout: VGPR r -> M = r + 8*khalf, N = lane%16. Coalesced 64B stores.
    #pragma unroll
    for (int r = 0; r < 8; ++r) {
        const int m = r + 8 * khalf;
        H[(size_t)(n0 + m) * UNITS + ncol] = acc[r];
    }
}

// ---------------------------------------------------------------------------
// Kernel 2: per-node attention logits. One wave per node; lane owns 4 units.
// a_tgt[n] = h[n,:] . KA[:128], a_src[n] = h[n,:] . KA[128:]
// ---------------------------------------------------------------------------
__global__ __launch_bounds__(256) void gat_logits(const float* __restrict__ H,
                                                  const float* __restrict__ KA,
                                                  float* __restrict__ a_tgt,
                                                  float* __restrict__ a_src)
{
    const int wave = threadIdx.x >> 5;
    const int lane = threadIdx.x & 31;
    const int n    = blockIdx.x * 8 + wave;
    if (n >= N_NODES) return;

    const float4 hv = *(const float4*)&H[(size_t)n * UNITS + lane * 4];
    const float4 kt = *(const float4*)&KA[lane * 4];
    const float4 ks = *(const float4*)&KA[UNITS + lane * 4];

    float st = hv.x * kt.x + hv.y * kt.y + hv.z * kt.z + hv.w * kt.w;
    float ss = hv.x * ks.x + hv.y * ks.y + hv.z * ks.z + hv.w * ks.w;

    #pragma unroll
    for (int off = 16; off > 0; off >>= 1) {
        st += __shfl_xor(st, off, 32);
        ss += __shfl_xor(ss, off, 32);
    }
    if (lane == 0) {
        a_tgt[n] = st;
        a_src[n] = ss;
    }
}

// ---------------------------------------------------------------------------
// Kernel 3: per-edge score + segment-sum denominator (1 atomic per edge).
// ---------------------------------------------------------------------------
__global__ __launch_bounds__(256) void gat_scores(const int* __restrict__ E,
                                                  const float* __restrict__ a_tgt,
                                                  const float* __restrict__ a_src,
                                                  float* __restrict__ scores,
                                                  float* __restrict__ denom)
{
    const int e = blockIdx.x * 256 + threadIdx.x;
    if (e >= N_EDGES) return;
    const int t = E[2 * e + 0];
    const int s = E[2 * e + 1];
    float x = a_tgt[t] + a_src[s];
    x = (x > 0.0f) ? x : NEG_SLOPE * x;           // leaky_relu
    x = fminf(2.0f, fmaxf(-2.0f, x));             // clip
    const float sc = expf(x);
    scores[e] = sc;
    atomicAdd(&denom[t], sc);
}

// ---------------------------------------------------------------------------
// Kernel 4: scatter out[tgt] += h[src] * attn. 128 threads per edge, 1 unit
// per thread. h (25.6 MB) is L2-resident on MI455X (192 MB L2), so the random
// gathers stay on-chip; atomics land in L2.
// ---------------------------------------------------------------------------
__global__ __launch_bounds__(256) void gat_scatter(const int* __restrict__ E,
                                                   const float* __restrict__ H,
                                                   const float* __restrict__ scores,
                                                   const float* __restrict__ denom,
                                                   float* __restrict__ out)
{
    const long long gid = (long long)blockIdx.x * 256 + threadIdx.x;
    const long long e   = gid >> 7;     // / 128
    const int       u   = (int)(gid & 127);
    if (e >= N_EDGES) return;
    const int t = E[2 * e + 0];
    const int s = E[2 * e + 1];
    const float attn = scores[e] / denom[t];
    atomicAdd(&out[(size_t)t * UNITS + u], H[(size_t)s * UNITS + u] * attn);
}

// ---------------------------------------------------------------------------
// Launch
// ---------------------------------------------------------------------------
extern "C" void kernel_launch(void* const* d_in, const int* in_sizes, int n_in,
                              void* d_out, int out_size, void* d_ws, size_t ws_size,
                              hipStream_t stream)
{
    const float* X  = (const float*)d_in[0];   // node_states [50000,256]
    const int*   E  = (const int*)  d_in[1];   // edges       [800000,2]
    const float* W  = (const float*)d_in[2];   // kernel      [256,128]
    const float* KA = (const float*)d_in[3];   // kernel_attention [256,1]
    float* out = (float*)d_out;                // [50000,128]

    // Workspace partition (256B aligned): H, a_tgt, a_src, denom, scores
    char*  ws  = (char*)d_ws;
    size_t off = 0;
    auto take = [&](size_t bytes) {
        char* p = ws + off;
        off = (off + bytes + 255) & ~(size_t)255;
        return p;
    };
    float* H      = (float*)take((size_t)N_NODES * UNITS * sizeof(float));
    float* a_tgt  = (float*)take((size_t)N_NODES * sizeof(float));
    float* a_src  = (float*)take((size_t)N_NODES * sizeof(float));
    float* denom  = (float*)take((size_t)N_NODES * sizeof(float));
    float* scores = (float*)take((size_t)N_EDGES * sizeof(float));

    hipMemsetAsync(denom, 0, (size_t)N_NODES * sizeof(float), stream);
    hipMemsetAsync(out,   0, (size_t)N_NODES * UNITS * sizeof(float), stream);

    gat_gemm_h <<<N_NODES / 16, 256, 0, stream>>>(X, W, H);
    gat_logits <<<(N_NODES + 7) / 8, 256, 0, stream>>>(H, KA, a_tgt, a_src);
    gat_scores <<<(N_EDGES + 255) / 256, 256, 0, stream>>>(E, a_tgt, a_src, scores, denom);

    const long long total = (long long)N_EDGES * UNITS;
    gat_scatter<<<(unsigned)((total + 255) / 256), 256, 0, stream>>>(E, H, scores, denom, out);
}